// erp2igd_encoder_10462540333217
// MI455X (gfx1250) — compile-verified
//
#include <hip/hip_runtime.h>

// D = A x B + C, f32 16x16x4 WMMA path on gfx1250 (wave32).
typedef __attribute__((ext_vector_type(2))) float v2f;
typedef __attribute__((ext_vector_type(4))) float v4f;
typedef __attribute__((ext_vector_type(8))) float v8f;

constexpr int       kC  = 16;
constexpr int       kK  = 8;
constexpr int       kQ  = 10;
constexpr long long kHW = 721LL * 1440LL;   // 1,038,240 elements per (b,c) slice
constexpr int       kN  = kQ * 256 * 256;   // 655,360 positions
constexpr int       kTiles = kN / 16;       // 40,960 16-row tiles
constexpr int       kWavesPerBlock = 8;     // 256 threads = 8 wave32

// One wave computes a 16(n) x 16(c) output tile for BOTH batches:
//   out[n, c] = relu( sum_k x[b, c, idx[n,k]] * w[c,k] + bias[c] )
// expressed as A(16x128) x blockdiag(W)(128x16) via 32 chained
// v_wmma_f32_16x16x4_f32. Indices / B-operand shared across the two batches.
__global__ __launch_bounds__(256)
void erp2igd_gather_dot_wmma(const float* __restrict__ x,
                             const float* __restrict__ w,
                             const float* __restrict__ bias,
                             const long long* __restrict__ idx,
                             float* __restrict__ out)
{
    const int lane = threadIdx.x & 31;
    const int L    = lane & 15;   // A-matrix row (n within tile) and D column (channel)
    const int hi   = lane >> 4;   // 0: K-slots {0,1}, 1: K-slots {2,3} of each WMMA
    const int tile = blockIdx.x * kWavesPerBlock + (threadIdx.x >> 5);
    const int n0   = tile * 16;

    // Per-lane gather indices: row n0+L, k in {hi*2, hi*2+1} and {4+hi*2, 4+hi*2+1}.
    // (lane-dependent addresses, compile-time register indexing afterwards)
    const long long* irow = idx + (long long)(n0 + L) * kK + hi * 2;
    const ulonglong2 e0 = *(const ulonglong2*)(irow);       // kbase 0
    const ulonglong2 e1 = *(const ulonglong2*)(irow + 4);   // kbase 4
    const unsigned i00 = (unsigned)e0.x, i01 = (unsigned)e0.y;
    const unsigned i10 = (unsigned)e1.x, i11 = (unsigned)e1.y;

    // Per-lane weights of channel L at the same k positions.
    const float2* pw = (const float2*)(w + L * kK + hi * 2);
    const float2 w0 = pw[0];   // k = hi*2, hi*2+1
    const float2 w1 = pw[2];   // k = 4+hi*2, 4+hi*2+1
    const float  bc = bias[L];

    v8f acc0 = {0.f,0.f,0.f,0.f,0.f,0.f,0.f,0.f};  // batch 0
    v8f acc1 = acc0;                               // batch 1

    const float* xb0 = x;
    const float* xb1 = x + (long long)kC * kHW;

#pragma unroll
    for (int g = 0; g < 32; ++g) {
        const int  cp  = g >> 1;      // source-channel block (compile-time)
        const bool par = (g & 1);     // kbase 0 / 4
        const unsigned ia = par ? i10 : i00;
        const unsigned ib = par ? i11 : i01;
        const float2   wv = par ? w1 : w0;

        const float* p0 = xb0 + (long long)cp * kHW;
        const float* p1 = xb1 + (long long)cp * kHW;

        // A operand: lane holds A[L, k0] / A[L, k0+1] (k0 = kbase + hi*2)
        v2f a0, a1, bm;
        a0.x = p0[ia]; a0.y = p0[ib];
        a1.x = p1[ia]; a1.y = p1[ib];

        // B operand: block-diagonal — only column cp is nonzero.
        const bool sel = (L == cp);
        bm.x = sel ? wv.x : 0.0f;
        bm.y = sel ? wv.y : 0.0f;

        acc0 = __builtin_amdgcn_wmma_f32_16x16x4_f32(
            false, a0, false, bm, (short)0, acc0, false, false);
        acc1 = __builtin_amdgcn_wmma_f32_16x16x4_f32(
            false, a1, false, bm, (short)0, acc1, false, false);
    }

    // Epilogue: bias + ReLU.
#pragma unroll
    for (int r = 0; r < 8; ++r) {
        acc0[r] = fmaxf(acc0[r] + bc, 0.0f);
        acc1[r] = fmaxf(acc1[r] + bc, 0.0f);
    }

    // D layout: lane L holds rows 0..7 (hi=0) / 8..15 (hi=1) of column c=L in
    // 8 consecutive VGPRs -> 8 consecutive n positions -> two b128 stores.
    const int q       = n0 >> 16;                 // 65536 = 256*256 per q-slice
    const int inslice = (n0 & 65535) + hi * 8;    // i*256 + j
    float* o0 = out + (((long long)(       q * kC + L)) << 16) + inslice;  // b=0
    float* o1 = out + (((long long)((kQ + q) * kC + L)) << 16) + inslice;  // b=1

    v4f s;
    s = __builtin_shufflevector(acc0, acc0, 0,1,2,3); *(v4f*)(o0)     = s;
    s = __builtin_shufflevector(acc0, acc0, 4,5,6,7); *(v4f*)(o0 + 4) = s;
    s = __builtin_shufflevector(acc1, acc1, 0,1,2,3); *(v4f*)(o1)     = s;
    s = __builtin_shufflevector(acc1, acc1, 4,5,6,7); *(v4f*)(o1 + 4) = s;
}

extern "C" void kernel_launch(void* const* d_in, const int* /*in_sizes*/, int /*n_in*/,
                              void* d_out, int /*out_size*/, void* /*d_ws*/, size_t /*ws_size*/,
                              hipStream_t stream) {
    const float*      x   = (const float*)d_in[0];
    const float*      w   = (const float*)d_in[1];
    const float*      b   = (const float*)d_in[2];
    const long long*  idx = (const long long*)d_in[3];
    float*            out = (float*)d_out;

    dim3 grid(kTiles / kWavesPerBlock);   // 5120 blocks
    dim3 block(256);                      // 8 wave32 per block
    hipLaunchKernelGGL(erp2igd_gather_dot_wmma, grid, block, 0, stream,
                       x, w, b, idx, out);
}